// NGCF_1056561954898
// MI455X (gfx1250) — compile-verified
//
#include <hip/hip_runtime.h>
#include <hip/hip_bf16.h>
#include <math.h>

typedef __attribute__((ext_vector_type(2))) float v2f;
typedef __attribute__((ext_vector_type(8))) float v8f;

#define N_USERS 100000
#define N_ITEMS 200000
#define N_TOT   300000
#define DIM     64

// ---------------------------------------------------------------------------
// init: ego = concat(user_emb, item_emb); acc (= d_out) = ego  (layer-0 term)
// ---------------------------------------------------------------------------
__global__ void ngcf_init(const float* __restrict__ u, const float* __restrict__ it,
                          float* __restrict__ ego, float* __restrict__ acc,
                          long long userElems, long long total) {
    long long i = (long long)blockIdx.x * blockDim.x + threadIdx.x;
    if (i >= total) return;
    float v = (i < userElems) ? u[i] : it[i - userElems];
    ego[i] = v;
    acc[i] = v;
}

__global__ void ngcf_zero(float* __restrict__ p, long long total) {
    long long i = (long long)blockIdx.x * blockDim.x + threadIdx.x;
    if (i < total) p[i] = 0.0f;
}

__global__ void ngcf_scale(float* __restrict__ p, float s, long long total) {
    long long i = (long long)blockIdx.x * blockDim.x + threadIdx.x;
    if (i < total) p[i] *= s;
}

// ---------------------------------------------------------------------------
// SpMM scatter: side[row] += val * ego[col].  One wave32 per edge; each lane
// owns 2 consecutive columns -> one b64 gather + 2 no-return f32 atomics.
// Edge scalars are wave-uniform: route the index through readfirstlane so the
// compiler can keep them on the scalar (SMEM/KMcnt) path.
// ---------------------------------------------------------------------------
__global__ void ngcf_spmm(const int* __restrict__ row, const int* __restrict__ col,
                          const float* __restrict__ val,
                          const float* __restrict__ x, float* __restrict__ y,
                          int nnz) {
    long long gid  = (long long)blockIdx.x * blockDim.x + threadIdx.x;
    long long edge = gid >> 5;
    int       lane = (int)(gid & 31);
    if (edge >= nnz) return;
    int es = __builtin_amdgcn_readfirstlane((int)edge);   // wave-uniform -> SGPR
    int   r = row[es];
    int   c = col[es];
    float v = val[es];
    const float2* src = (const float2*)(x + (long long)c * DIM) + lane;
    float2 xv = *src;
    float* dst = y + (long long)r * DIM + 2 * lane;
    atomicAdd(dst + 0, v * xv.x);
    atomicAdd(dst + 1, v * xv.y);
}

// ---------------------------------------------------------------------------
// Fused layer:
//   pre  = side @ Wgc + bgc + (ego*side) @ Wbi + bbi      (two WMMA chains,
//          one shared f32 accumulator, V_WMMA_F32_16X16X4_F32, K=64 -> 16 steps)
//   ego' = leaky_relu(pre, 0.2)                            (written in place)
//   acc += ego' / max(||ego'||_2, eps)                     (row L2 normalize)
// Block = 128 threads (4 waves) on a 16-row tile; wave w owns columns
// [16w, 16w+16).  B fragments (32 weight floats per lane per matrix) are
// preloaded into VGPRs so the MMA loop is only LDS A-loads + WMMA.
// ---------------------------------------------------------------------------
__global__ void __launch_bounds__(128)
ngcf_layer(float* __restrict__ ego, const float* __restrict__ side,
           const float* __restrict__ Wgc, const float* __restrict__ bgc,
           const float* __restrict__ Wbi, const float* __restrict__ bbi,
           float* __restrict__ acc) {
    __shared__ float s_side[16 * DIM];
    __shared__ float s_bi[16 * DIM];
    __shared__ float s_out[16 * DIM];
    __shared__ float s_rnorm[16];

    const int       tid  = threadIdx.x;
    const long long base = (long long)blockIdx.x * 16 * DIM;

    // Stage the 16x64 side tile and the elementwise ego*side tile in LDS.
    #pragma unroll
    for (int i = tid; i < 16 * DIM; i += 128) {
        float sv = side[base + i];
        s_side[i] = sv;
        s_bi[i]   = ego[base + i] * sv;
    }

    const int wave = tid >> 5;          // 0..3 : column tile
    const int lane = tid & 31;
    const int nloc = lane & 15;         // column within tile / A-row M
    const int half = lane >> 4;         // selects K pair {0,1} vs {2,3}
    const int n    = 16 * wave + nloc;  // global column 0..63
    const int m    = nloc;              // A-matrix row for this lane

    // Preload all B fragments (weights are block-shared, WGP$-resident).
    v2f bg[16], bb[16];
    #pragma unroll
    for (int kk = 0; kk < 16; ++kk) {
        const int kb = 4 * kk + 2 * half;
        bg[kk].x = Wgc[kb * DIM + n];
        bg[kk].y = Wgc[(kb + 1) * DIM + n];
        bb[kk].x = Wbi[kb * DIM + n];
        bb[kk].y = Wbi[(kb + 1) * DIM + n];
    }

    __syncthreads();

    v8f c = {};                         // f32 16x16 accumulator (8 VGPRs)
    #pragma unroll
    for (int kk = 0; kk < 16; ++kk) {
        const int kb = 4 * kk + 2 * half;   // this lane's K base per ISA A layout
        v2f a_gc, a_bi;
        a_gc.x = s_side[m * DIM + kb];
        a_gc.y = s_side[m * DIM + kb + 1];
        a_bi.x = s_bi[m * DIM + kb];
        a_bi.y = s_bi[m * DIM + kb + 1];
        // D = A*B + C, exact f32 matrix core path
        c = __builtin_amdgcn_wmma_f32_16x16x4_f32(false, a_gc, false, bg[kk],
                                                  (short)0, c, false, false);
        c = __builtin_amdgcn_wmma_f32_16x16x4_f32(false, a_bi, false, bb[kk],
                                                  (short)0, c, false, false);
    }

    // Fused bias + leaky_relu(0.2); stash activated tile in LDS.
    const float bias = bgc[n] + bbi[n];
    #pragma unroll
    for (int r = 0; r < 8; ++r) {
        const int mr = r + 8 * half;    // C/D layout: VGPR r holds row r (+8 for hi lanes)
        float x = c[r] + bias;
        x = (x > 0.0f) ? x : 0.2f * x;
        s_out[mr * DIM + n] = x;
    }
    __syncthreads();

    // Row L2 norms (16 rows).
    if (tid < 16) {
        float s = 0.0f;
        #pragma unroll
        for (int d = 0; d < DIM; ++d) {
            float v = s_out[tid * DIM + d];
            s += v * v;
        }
        s_rnorm[tid] = 1.0f / fmaxf(sqrtf(s), 1e-12f);
    }
    __syncthreads();

    // Write new ego (in place, block-private rows) and accumulate normalized.
    #pragma unroll
    for (int i = tid; i < 16 * DIM; i += 128) {
        const int mr = i >> 6;
        float v = s_out[i];
        ego[base + i] = v;
        acc[base + i] += v * s_rnorm[mr];
    }
}

// ---------------------------------------------------------------------------
extern "C" void kernel_launch(void* const* d_in, const int* in_sizes, int n_in,
                              void* d_out, int out_size, void* d_ws, size_t ws_size,
                              hipStream_t stream) {
    const float* user_emb = (const float*)d_in[0];
    const float* item_emb = (const float*)d_in[1];
    const float* W_gc     = (const float*)d_in[2];   // [3,64,64]
    const float* b_gc     = (const float*)d_in[3];   // [3,1,64]
    const float* W_bi     = (const float*)d_in[4];
    const float* b_bi     = (const float*)d_in[5];
    const float* adj_val  = (const float*)d_in[6];
    const int*   adj_row  = (const int*)d_in[7];
    const int*   adj_col  = (const int*)d_in[8];
    float*       out      = (float*)d_out;           // [N_TOT, 64] mean accumulator

    const long long ND   = (long long)N_TOT * DIM;   // 19.2M
    const long long NUel = (long long)N_USERS * DIM;
    const int nnz = in_sizes[6];

    float* ego  = (float*)d_ws;          // [N_TOT*64]
    float* side = ego + ND;              // [N_TOT*64]

    const int threads = 256;
    const int gridND  = (int)((ND + threads - 1) / threads);

    ngcf_init<<<gridND, threads, 0, stream>>>(user_emb, item_emb, ego, out, NUel, ND);

    const long long spmmThreads = (long long)nnz * 32;
    const int gridSpmm = (int)((spmmThreads + threads - 1) / threads);

    for (int k = 0; k < 3; ++k) {
        ngcf_zero<<<gridND, threads, 0, stream>>>(side, ND);
        ngcf_spmm<<<gridSpmm, threads, 0, stream>>>(adj_row, adj_col, adj_val,
                                                    ego, side, nnz);
        ngcf_layer<<<N_TOT / 16, 128, 0, stream>>>(ego, side,
                                                   W_gc + (long long)k * DIM * DIM,
                                                   b_gc + (long long)k * DIM,
                                                   W_bi + (long long)k * DIM * DIM,
                                                   b_bi + (long long)k * DIM,
                                                   out);
    }
    ngcf_scale<<<gridND, threads, 0, stream>>>(out, 0.25f, ND);
}